// recurrent_LRGCN_54202487275556
// MI455X (gfx1250) — compile-verified
//
#include <hip/hip_runtime.h>
#include <hip/hip_bf16.h>
#include <math.h>

typedef float v2f __attribute__((ext_vector_type(2)));
typedef float v8f __attribute__((ext_vector_type(8)));

#define N_PAD   3000
#define N_REAL  2500
#define F_IN    1024
#define H1      256
#define OUT_D   128
#define N_EDGE  300000

// ---------------------------------------------------------------------------
// K1: xh = relu(x_pad @ fc0_w^T + fc0_b)   [3000,1024]x[1024,256] fp32 WMMA
// One wave computes a 16x64 tile (4 x v8f accumulators), reusing the A fetch.
// A 16x4 f32 layout: lane[0:16) M=lane K={k,k+1}; lane[16:32) M=lane-16 K={k+2,k+3}
// Boundary rows handled by pointer clamp + 0/1 mask multiply (no exec
// divergence inside the k-loop).
// ---------------------------------------------------------------------------
__global__ __launch_bounds__(128) void k_gemm_xh(const float* __restrict__ x,
                                                 const float* __restrict__ w0,
                                                 const float* __restrict__ b0,
                                                 float* __restrict__ xh) {
  const int lane = threadIdx.x & 31;
  const int wave = threadIdx.x >> 5;
  const int half = lane >> 4;
  const int l15  = lane & 15;
  const int m0   = blockIdx.x * 16;
  const int m    = m0 + l15;                    // A-matrix row owned by this lane
  const int mc   = (m < N_REAL) ? m : 0;        // clamped (valid) row pointer
  const float amask = (m < N_REAL) ? 1.f : 0.f; // zero-out pad rows branchlessly
  const int nb   = wave * 64;                   // this wave's N range

  v8f c0 = {}, c1 = {}, c2 = {}, c3 = {};
  const float* arow  = x  + (size_t)mc * F_IN;
  const float* brow0 = w0 + (size_t)(nb +  0 + l15) * F_IN;
  const float* brow1 = w0 + (size_t)(nb + 16 + l15) * F_IN;
  const float* brow2 = w0 + (size_t)(nb + 32 + l15) * F_IN;
  const float* brow3 = w0 + (size_t)(nb + 48 + l15) * F_IN;

  #pragma unroll 2
  for (int k0 = 0; k0 < F_IN; k0 += 4) {
    const int kk = k0 + 2 * half;
    v2f a = *(const v2f*)(arow + kk);
    a.x *= amask; a.y *= amask;
    v2f b0v = *(const v2f*)(brow0 + kk);
    v2f b1v = *(const v2f*)(brow1 + kk);
    v2f b2v = *(const v2f*)(brow2 + kk);
    v2f b3v = *(const v2f*)(brow3 + kk);
    c0 = __builtin_amdgcn_wmma_f32_16x16x4_f32(false, a, false, b0v, (short)0, c0, false, false);
    c1 = __builtin_amdgcn_wmma_f32_16x16x4_f32(false, a, false, b1v, (short)0, c1, false, false);
    c2 = __builtin_amdgcn_wmma_f32_16x16x4_f32(false, a, false, b2v, (short)0, c2, false, false);
    c3 = __builtin_amdgcn_wmma_f32_16x16x4_f32(false, a, false, b3v, (short)0, c3, false, false);
  }

  v8f cs[4] = {c0, c1, c2, c3};
  #pragma unroll
  for (int t = 0; t < 4; ++t) {
    const int col  = nb + t * 16 + l15;
    const float bias = b0[col];
    #pragma unroll
    for (int v = 0; v < 8; ++v) {
      const int row = m0 + v + 8 * half;        // C/D layout: VGPR v -> M=v / v+8
      if (row < N_PAD) {
        float val = cs[t][v] + bias;            // pad rows: 0*W + b -> relu(b)
        xh[(size_t)row * H1 + col] = val > 0.f ? val : 0.f;
      }
    }
  }
}

// ---------------------------------------------------------------------------
// K2: zero agg[3000*256] and cnt[3000]
// ---------------------------------------------------------------------------
__global__ void k_zero(float* __restrict__ agg, float* __restrict__ cnt) {
  const int i = blockIdx.x * blockDim.x + threadIdx.x;
  if (i < N_PAD * H1) agg[i] = 0.f;
  if (i < N_PAD)      cnt[i] = 0.f;
}

// ---------------------------------------------------------------------------
// K3: edge scatter-sum: agg[dst] += xh[src]; cnt[dst] += 1. One wave per edge,
// lane-coalesced f32 atomics (all traffic stays in the 192MB L2).
// ---------------------------------------------------------------------------
__global__ __launch_bounds__(256) void k_scatter(const int* __restrict__ ei,
                                                 const float* __restrict__ xh,
                                                 float* __restrict__ agg,
                                                 float* __restrict__ cnt) {
  const int wave = threadIdx.x >> 5;
  const int lane = threadIdx.x & 31;
  const int e = blockIdx.x * 8 + wave;
  if (e >= N_EDGE) return;
  const int s = ei[e];
  const int d = ei[N_EDGE + e];
  const float* sr = xh  + (size_t)s * H1;
  float*       dr = agg + (size_t)d * H1;
  if (lane == 0) atomicAdd(&cnt[d], 1.0f);
  #pragma unroll
  for (int j = 0; j < H1 / 32; ++j)
    atomicAdd(&dr[lane + 32 * j], sr[lane + 32 * j]);
}

// ---------------------------------------------------------------------------
// K4: gates + fused LSTM cell.
//   g[gate] = (agg/deg) @ Wx_rel[gate] + xh @ Wx_root[gate] + bx + bh
//   (H_prev = C_prev = 0 so Wh terms vanish; F-gate multiplies zero)
// Block = 4 waves; wave g computes its gate's 16x128 slab with fp32 WMMA.
// The segment-mean 1/deg scale is folded into the agg A-operand (it also
// serves as the boundary mask); xh gets an explicit 0/1 mask. No exec
// divergence in the k-loop. Gates exchanged via LDS, fused elementwise -> H.
// ---------------------------------------------------------------------------
__global__ __launch_bounds__(128) void k_gates(const float* __restrict__ xh,
                                               const float* __restrict__ agg,
                                               const float* __restrict__ cnt,
                                               const float* __restrict__ Wrel,
                                               const float* __restrict__ Wroot,
                                               const float* __restrict__ bx,
                                               const float* __restrict__ bh,
                                               float* __restrict__ Hout) {
  __shared__ float gbuf[4][16][OUT_D];
  const int lane = threadIdx.x & 31;
  const int g    = threadIdx.x >> 5;   // gate index: i,f,c,o
  const int half = lane >> 4;
  const int l15  = lane & 15;
  const int m0   = blockIdx.x * 16;
  const int m    = m0 + l15;
  const int mc   = (m < N_PAD) ? m : (N_PAD - 1);

  float inv = 0.f, xmask = 0.f;        // inv==0 also masks pad rows of agg
  if (m < N_PAD) {
    const float c = cnt[m];
    inv = 1.0f / (c > 1.0f ? c : 1.0f);
    xmask = 1.0f;
  }
  const float* aggr  = agg + (size_t)mc * H1;
  const float* xhr   = xh  + (size_t)mc * H1;
  const float* wrel  = Wrel  + (size_t)g * H1 * OUT_D;
  const float* wroot = Wroot + (size_t)g * H1 * OUT_D;

  v8f acc[8] = {};
  for (int k0 = 0; k0 < H1; k0 += 4) {
    const int kk = k0 + 2 * half;
    v2f aa = *(const v2f*)(aggr + kk);
    aa.x *= inv; aa.y *= inv;                  // segment-mean + boundary mask
    v2f ax = *(const v2f*)(xhr + kk);
    ax.x *= xmask; ax.y *= xmask;
    #pragma unroll
    for (int t = 0; t < 8; ++t) {
      const int o = t * 16 + l15;
      v2f br = { wrel [(size_t)kk * OUT_D + o], wrel [(size_t)(kk + 1) * OUT_D + o] };
      v2f bo = { wroot[(size_t)kk * OUT_D + o], wroot[(size_t)(kk + 1) * OUT_D + o] };
      acc[t] = __builtin_amdgcn_wmma_f32_16x16x4_f32(false, aa, false, br, (short)0, acc[t], false, false);
      acc[t] = __builtin_amdgcn_wmma_f32_16x16x4_f32(false, ax, false, bo, (short)0, acc[t], false, false);
    }
  }

  #pragma unroll
  for (int t = 0; t < 8; ++t) {
    const int o = t * 16 + l15;
    const float bias = bx[g * OUT_D + o] + bh[g * OUT_D + o];
    #pragma unroll
    for (int v = 0; v < 8; ++v)
      gbuf[g][v + 8 * half][o] = acc[t][v] + bias;
  }
  __syncthreads();

  for (int idx = threadIdx.x; idx < 16 * OUT_D; idx += 128) {
    const int r = idx >> 7;
    const int o = idx & (OUT_D - 1);
    const int row = m0 + r;
    if (row < N_PAD) {
      const float gi = gbuf[0][r][o];
      const float gc = gbuf[2][r][o];
      const float go = gbuf[3][r][o];
      const float I = 1.f / (1.f + __expf(-gi));
      const float T = tanhf(gc);
      const float O = 1.f / (1.f + __expf(-go));
      const float C = I * T;                 // F*C_prev == 0 (C_prev zero-init)
      Hout[(size_t)row * OUT_D + o] = O * tanhf(C);
    }
  }
}

// ---------------------------------------------------------------------------
// K5: out[n] = H[n] . fc_w + fc_b   for n < 2500
// ---------------------------------------------------------------------------
__global__ void k_out(const float* __restrict__ H, const float* __restrict__ fw,
                      const float* __restrict__ fb, float* __restrict__ out) {
  const int n = blockIdx.x * blockDim.x + threadIdx.x;
  if (n >= N_REAL) return;
  const float* hr = H + (size_t)n * OUT_D;
  float s = 0.f;
  #pragma unroll 4
  for (int o = 0; o < OUT_D; ++o) s += hr[o] * fw[o];
  out[n] = s + fb[0];
}

extern "C" void kernel_launch(void* const* d_in, const int* in_sizes, int n_in,
                              void* d_out, int out_size, void* d_ws, size_t ws_size,
                              hipStream_t stream) {
  (void)in_sizes; (void)n_in; (void)out_size; (void)ws_size;
  const float* x      = (const float*)d_in[0];
  const int*   ei     = (const int*)  d_in[1];
  const float* fc0_w  = (const float*)d_in[2];
  const float* fc0_b  = (const float*)d_in[3];
  const float* Wx_rel = (const float*)d_in[4];
  const float* Wx_root= (const float*)d_in[5];
  const float* bx     = (const float*)d_in[6];
  // d_in[7]/d_in[8] (Wh_rel/Wh_root) multiply the zero hidden state -> dead
  const float* bh     = (const float*)d_in[9];
  const float* fc_w   = (const float*)d_in[10];
  const float* fc_b   = (const float*)d_in[11];

  float* xh  = (float*)d_ws;                   // [3000,256]
  float* agg = xh  + (size_t)N_PAD * H1;       // [3000,256]
  float* cnt = agg + (size_t)N_PAD * H1;       // [3000]

  float* out  = (float*)d_out;                 // [2500]
  float* Hout = out + N_REAL;                  // [3000,128]

  k_gemm_xh<<<dim3((N_PAD + 15) / 16), dim3(128), 0, stream>>>(x, fc0_w, fc0_b, xh);
  k_zero   <<<dim3((N_PAD * H1 + 255) / 256), dim3(256), 0, stream>>>(agg, cnt);
  k_scatter<<<dim3((N_EDGE + 7) / 8), dim3(256), 0, stream>>>(ei, xh, agg, cnt);
  k_gates  <<<dim3((N_PAD + 15) / 16), dim3(128), 0, stream>>>(xh, agg, cnt, Wx_rel, Wx_root, bx, bh, Hout);
  k_out    <<<dim3((N_REAL + 255) / 256), dim3(256), 0, stream>>>(Hout, fc_w, fc_b, out);
}